// FocalLoss_64871186039080
// MI455X (gfx1250) — compile-verified
//
#include <hip/hip_runtime.h>
#include <hip/hip_bf16.h>

// ---------------------------------------------------------------------------
// RetinaNet focal loss for MI455X (gfx1250, wave32).
// Phase 1: per-anchor IoU argmax / pos assignment / smooth-L1 reg partials.
// Phase 2: dense negative focal term over [B,A,C]; b128 streaming loads,
//          raw v_log_f32 (__builtin_amdgcn_logf), folded into
//          V_WMMA_F32_16X16X4_F32 (ones x data + C) as an exact fp32 reducer.
// Phase 3: deterministic finalize (fixed-order double reduction).
// ---------------------------------------------------------------------------

typedef __attribute__((ext_vector_type(2))) float v2f;
typedef __attribute__((ext_vector_type(8))) float v8f;

#define ALPHA_F 0.25f
#define LN2_F 0.69314718055994531f

__device__ __forceinline__ float clampc(float x) {
    return fminf(fmaxf(x, 1e-4f), 1.0f - 1e-4f);
}

// log(x) for x > 0: raw v_log_f32 (log2) * ln2 -- no denorm/inf fixups
__device__ __forceinline__ float fast_ln(float x) {
#if __has_builtin(__builtin_amdgcn_logf)
    return LN2_F * __builtin_amdgcn_logf(x);
#else
    return __logf(x);
#endif
}

// focal term for targets == 0 (applies to every (a,c) element)
__device__ __forceinline__ float neg_term(float x) {
    x = clampc(x);
#if __has_builtin(__builtin_amdgcn_logf)
    // 0.75 * x^2 * (-log(1-x)) ; fold -0.75*ln2 into one constant
    return (-0.75f * LN2_F) * (x * x) * __builtin_amdgcn_logf(1.0f - x);
#else
    return 0.75f * x * x * (-__logf(1.0f - x));
#endif
}

// focal term for targets == 1 (target class of a positive anchor)
__device__ __forceinline__ float pos_term(float x) {
    x = clampc(x);
    float om = 1.0f - x;
#if __has_builtin(__builtin_amdgcn_logf)
    return (-ALPHA_F * LN2_F) * (om * om) * __builtin_amdgcn_logf(x);
#else
    return ALPHA_F * om * om * (-__logf(x));
#endif
}

// ---------------------------------------------------------------------------
// Phase 2: sum of neg_term over all A*C elements of one sample.
// grid = (BLOCKS2, B), block = 256 (8 waves). Each wave consumes full chunks
// of 128 elements (float4 per lane) with no bounds checks; the <128-element
// residue is handled by wave 0 of each sample and folded into the reduction.
// ---------------------------------------------------------------------------
__global__ __launch_bounds__(256) void focal_neg_sum(
    const float* __restrict__ cls, float* __restrict__ p2, int N, int nfull) {
    const int b    = blockIdx.y;
    const int lane = threadIdx.x & 31;
    const int wg   = blockIdx.x * (blockDim.x >> 5) + (threadIdx.x >> 5);
    const int W    = gridDim.x * (blockDim.x >> 5);
    const float* __restrict__ base = cls + (size_t)b * (size_t)N;

    v8f acc = {0.f, 0.f, 0.f, 0.f, 0.f, 0.f, 0.f, 0.f};
    const v2f ones = {1.0f, 1.0f};
    const size_t stepElems = (size_t)W * 128;

    for (int chunk = wg; chunk < nfull; chunk += W) {
        const size_t idx = (size_t)chunk * 128 + (size_t)lane * 4;
        const float4 x = *(const float4*)(base + idx);
        __builtin_prefetch(base + idx + stepElems, 0, 0);  // next chunk of this wave
        v2f b0, b1;
        b0[0] = neg_term(x.x);
        b0[1] = neg_term(x.y);
        b1[0] = neg_term(x.z);
        b1[1] = neg_term(x.w);
#if __has_builtin(__builtin_amdgcn_wmma_f32_16x16x4_f32)
        // D = ones(16x4) * B(4x16) + C : every D[m][n] accumulates colsum[n];
        // exact fp32 adds, A==ones makes the lane->(k,n) mapping irrelevant.
        acc = __builtin_amdgcn_wmma_f32_16x16x4_f32(
            false, ones, false, b0, (short)0, acc, false, false);
        acc = __builtin_amdgcn_wmma_f32_16x16x4_f32(
            false, ones, false, b1, (short)0, acc, false, false);
#else
        acc[0] += (b0[0] + b0[1]) + (b1[0] + b1[1]);
#endif
    }

    // residue elements [nfull*128, N): wave-uniform branch, wave 0 only
    float extra = 0.0f;
    if (wg == 0) {
        for (int i = nfull * 128 + lane; i < N; i += 32) extra += neg_term(base[i]);
    }

    // lanes 0-15 hold D[0][n], lanes 16-31 hold D[8][n] (identical rows):
    // 32-lane sum of acc[0] == 2 * wmma total, so pre-scale by 0.5.
#if __has_builtin(__builtin_amdgcn_wmma_f32_16x16x4_f32)
    float v = 0.5f * acc[0] + extra;
#else
    float v = acc[0] + extra;
#endif
    for (int off = 16; off; off >>= 1) v += __shfl_xor(v, off, 32);
    if (lane == 0) p2[(size_t)b * (size_t)W + wg] = v;
}

// ---------------------------------------------------------------------------
// Phase 1: per-anchor assignment + positive-anchor partials.
// grid = (ceil(A/256), B), block = 256. GT boxes/labels/areas cached in LDS.
// ---------------------------------------------------------------------------
__global__ __launch_bounds__(256) void anchor_assign(
    const float* __restrict__ boxes, const int* __restrict__ labels,
    const float* __restrict__ reg, const float* __restrict__ cls,
    const float* __restrict__ anch, float* __restrict__ p1c,
    float* __restrict__ p1r, int* __restrict__ p1n, int A_, int C_, int M_) {
    __shared__ float sb[4 * 96];
    __shared__ float sarea[96];
    __shared__ int sl[96];
    __shared__ float rc[256];
    __shared__ float rr[256];
    __shared__ int rn[256];

    const int b = blockIdx.y;
    const int tid = threadIdx.x;

    for (int m = tid; m < M_; m += blockDim.x) {
        const float* bp = boxes + ((size_t)b * M_ + m) * 4;
        const float bx1 = bp[0], by1 = bp[1], bx2 = bp[2], by2 = bp[3];
        sb[m * 4 + 0] = bx1;
        sb[m * 4 + 1] = by1;
        sb[m * 4 + 2] = bx2;
        sb[m * 4 + 3] = by2;
        sarea[m] = (bx2 - bx1) * (by2 - by1);
        sl[m] = labels[(size_t)b * M_ + m];
    }
    __syncthreads();

    float corr = 0.0f, regc = 0.0f;
    int np = 0;
    const int a = blockIdx.x * blockDim.x + tid;
    if (a < A_) {
        const float* ap = anch + (size_t)a * 4;  // (y1,x1,y2,x2)
        const float ay1 = ap[0], ax1 = ap[1], ay2 = ap[2], ax2 = ap[3];
        const float aw = ax2 - ax1, ah = ay2 - ay1;
        const float aarea = aw * ah;

        float best = -2.0f;
        int bm = 0;
        for (int m = 0; m < M_; ++m) {
            const float bx1 = sb[m * 4 + 0], by1 = sb[m * 4 + 1];
            const float bx2 = sb[m * 4 + 2], by2 = sb[m * 4 + 3];
            float iw = fminf(ax2, bx2) - fmaxf(ax1, bx1);
            float ih = fminf(ay2, by2) - fmaxf(ay1, by1);
            iw = fmaxf(iw, 0.0f);
            ih = fmaxf(ih, 0.0f);
            const float inter = iw * ih;
            const float ua = fmaxf(aarea + sarea[m] - inter, 1e-8f);
            float iou = inter / ua;
            iou = (sl[m] != 0) ? iou : -1.0f;  // mask padded GT
            if (iou > best) {  // strict > keeps first index (jnp.argmax)
                best = iou;
                bm = m;
            }
        }

        const float gx1 = sb[bm * 4 + 0], gy1 = sb[bm * 4 + 1];
        const float gx2 = sb[bm * 4 + 2], gy2 = sb[bm * 4 + 3];
        const int lr = sl[bm];
        const int alab = (lr != 0) ? (lr - 1) : 0;
        const float gw0 = gx2 - gx1, gh0 = gy2 - gy1;
        const bool big = gw0 * gh0 > 100.0f;
        const bool pos = big ? (best >= 0.5f) : (best >= 0.15f);

        if (pos) {
            np = 1;
            // classification correction at the target class
            const float c = cls[((size_t)b * A_ + a) * C_ + alab];
            corr = pos_term(c) - neg_term(c);
            // regression smooth-L1
            const float gcx = gx1 + 0.5f * gw0, gcy = gy1 + 0.5f * gh0;
            const float gw = fmaxf(gw0, 1.0f), gh = fmaxf(gh0, 1.0f);
            const float acx = ax1 + 0.5f * aw, acy = ay1 + 0.5f * ah;
            float t[4];
            t[0] = (gcy - acy) / ah;     // tdy
            t[1] = (gcx - acx) / aw;     // tdx
            t[2] = fast_ln(gh / ah);     // tdh
            t[3] = fast_ln(gw / aw);     // tdw
            const float* rp = reg + ((size_t)b * A_ + a) * 4;
#pragma unroll
            for (int k = 0; k < 4; ++k) {
                const float d = fabsf(t[k] - rp[k]);
                regc += (d <= 1.0f / 9.0f) ? 4.5f * d * d : (d - 0.5f / 9.0f);
            }
        }
    }

    rc[tid] = corr;
    rr[tid] = regc;
    rn[tid] = np;
    __syncthreads();
    for (int s = 128; s; s >>= 1) {
        if (tid < s) {
            rc[tid] += rc[tid + s];
            rr[tid] += rr[tid + s];
            rn[tid] += rn[tid + s];
        }
        __syncthreads();
    }
    if (tid == 0) {
        const int o = b * gridDim.x + blockIdx.x;
        p1c[o] = rc[0];
        p1r[o] = rr[0];
        p1n[o] = rn[0];
    }
}

// ---------------------------------------------------------------------------
// Phase 3: deterministic finalize. Thread b reduces sample b's partials in a
// fixed order (double), then thread 0 averages across the batch.
// ---------------------------------------------------------------------------
__global__ void focal_finalize(const float* __restrict__ p2,
                               const float* __restrict__ p1c,
                               const float* __restrict__ p1r,
                               const int* __restrict__ p1n,
                               float* __restrict__ out, int W, int G1, int B_) {
    __shared__ double scl[32];
    __shared__ double srg[32];
    const int b = threadIdx.x;
    if (b < B_) {
        double neg = 0.0;
        const float* q = p2 + (size_t)b * W;
        for (int k = 0; k < W; ++k) neg += (double)q[k];
        double corr = 0.0, regs = 0.0;
        long long np = 0;
        const float* c = p1c + (size_t)b * G1;
        const float* r = p1r + (size_t)b * G1;
        const int* n = p1n + (size_t)b * G1;
        for (int k = 0; k < G1; ++k) {
            corr += (double)c[k];
            regs += (double)r[k];
            np += n[k];
        }
        const double dn = (np > 0) ? (double)np : 1.0;
        scl[b] = (neg + corr) / dn;
        srg[b] = (np > 0) ? regs / ((double)np * 4.0) : 0.0;
    }
    __syncthreads();
    if (b == 0) {
        double s1 = 0.0, s2 = 0.0;
        for (int k = 0; k < B_; ++k) {
            s1 += scl[k];
            s2 += srg[k];
        }
        out[0] = (float)(s1 / (double)B_);
        out[1] = (float)(s2 / (double)B_ * 50.0);
    }
}

extern "C" void kernel_launch(void* const* d_in, const int* in_sizes, int n_in,
                              void* d_out, int out_size, void* d_ws,
                              size_t ws_size, hipStream_t stream) {
    (void)in_sizes; (void)n_in; (void)out_size; (void)ws_size;
    const int B_ = 8, A_ = 49104, M_ = 90, C_ = 90;

    const float* boxes = (const float*)d_in[0];   // [B,M,4] f32 (x1,y1,x2,y2)
    const int* labels  = (const int*)d_in[1];     // [B,M]   i32 (0 = padding)
    const float* reg   = (const float*)d_in[2];   // [B,A,4] f32
    const float* cls   = (const float*)d_in[3];   // [B,A,C] f32
    const float* anch  = (const float*)d_in[4];   // [1,A,4] f32 (y1,x1,y2,x2)
    float* out = (float*)d_out;                   // {cls_loss, det_loss}

    const int G1 = (A_ + 255) / 256;              // 192 blocks per sample
    const int BLOCKS2 = 256;                      // phase-2 blocks per sample
    const int W = BLOCKS2 * 8;                    // waves per sample (2048)
    const int N = A_ * C_;                        // 4,419,360
    const int nfull = N / 128;                    // full 128-elem wave chunks

    // Workspace layout (all slots fully rewritten every launch; no init pass)
    float* p2  = (float*)d_ws;                    // [B*W]
    float* p1c = p2 + (size_t)B_ * W;             // [B*G1]
    float* p1r = p1c + (size_t)B_ * G1;           // [B*G1]
    int*   p1n = (int*)(p1r + (size_t)B_ * G1);   // [B*G1]

    anchor_assign<<<dim3(G1, B_), 256, 0, stream>>>(boxes, labels, reg, cls,
                                                    anch, p1c, p1r, p1n, A_,
                                                    C_, M_);
    focal_neg_sum<<<dim3(BLOCKS2, B_), 256, 0, stream>>>(cls, p2, N, nfull);
    focal_finalize<<<1, 32, 0, stream>>>(p2, p1c, p1r, p1n, out, W, G1, B_);
}